// PME_15333033247244
// MI455X (gfx1250) — compile-verified
//
#include <hip/hip_runtime.h>
#include <math.h>

// ---------------------------------------------------------------------------
// PME energy for MI455X (gfx1250). Reciprocal-space FFT done as DFT-GEMMs
// using V_WMMA_F32_16X16X4_F32. Everything fits in L2 (192MB); job is
// latency-bound, so f32 WMMA (precision) over f8/f16 (throughput).
// ---------------------------------------------------------------------------

typedef __attribute__((ext_vector_type(2))) float v2f;
typedef __attribute__((ext_vector_type(8))) float v8f;

#define NATOMS   4096
#define K1D      64
#define KTOT     (64*64*64)
#define CUT2     81.0f
#define ALPHA_C  0.34f
#define COULOMB_C 138.935456f
#define PI_F     3.14159265358979f

__device__ inline float det3(const float* b) {
    return b[0]*(b[4]*b[8]-b[5]*b[7])
         - b[1]*(b[3]*b[8]-b[5]*b[6])
         + b[2]*(b[3]*b[7]-b[4]*b[6]);
}

__device__ inline void inv3(const float* b, float* inv) {
    float d  = det3(b);
    float id = 1.0f / d;
    inv[0] =  (b[4]*b[8]-b[5]*b[7])*id;
    inv[3] = -(b[3]*b[8]-b[5]*b[6])*id;
    inv[6] =  (b[3]*b[7]-b[4]*b[6])*id;
    inv[1] = -(b[1]*b[8]-b[2]*b[7])*id;
    inv[4] =  (b[0]*b[8]-b[2]*b[6])*id;
    inv[7] = -(b[0]*b[7]-b[1]*b[6])*id;
    inv[2] =  (b[1]*b[5]-b[2]*b[4])*id;
    inv[5] = -(b[0]*b[5]-b[2]*b[3])*id;
    inv[8] =  (b[0]*b[4]-b[1]*b[3])*id;
}

// min-image: delta (row vector) -> wrapped delta, general triclinic box
__device__ inline void min_image(float& dx, float& dy, float& dz,
                                 const float* bx, const float* ib) {
    float sx = dx*ib[0] + dy*ib[3] + dz*ib[6];
    float sy = dx*ib[1] + dy*ib[4] + dz*ib[7];
    float sz = dx*ib[2] + dy*ib[5] + dz*ib[8];
    sx -= rintf(sx); sy -= rintf(sy); sz -= rintf(sz);
    dx = sx*bx[0] + sy*bx[3] + sz*bx[6];
    dy = sx*bx[1] + sy*bx[4] + sz*bx[7];
    dz = sx*bx[2] + sy*bx[5] + sz*bx[8];
}

// order-4 cardinal B-spline weights (OpenMM convention), t in [0,1)
__device__ inline void bsp4(float t, float* w) {
    float w20 = 1.0f - t, w21 = t;
    float w30 = 0.5f*(1.0f - t)*w20;
    float w31 = 0.5f*((t + 1.0f)*w20 + (2.0f - t)*w21);
    float w32 = 0.5f*t*w21;
    const float d = 1.0f/3.0f;
    w[0] = d*(1.0f - t)*w30;
    w[1] = d*((t + 2.0f)*w30 + (2.0f - t)*w31);
    w[2] = d*((t + 1.0f)*w31 + (3.0f - t)*w32);
    w[3] = d*t*w32;
}

// |b(m)|^-2 for order-4, K=64: FFT of [1/6, 2/3, 1/6] at index m
__device__ inline float bmodf(int m) {
    float th = (2.0f*PI_F/64.0f)*(float)m;
    float re = 0.16666667f + 0.66666667f*__cosf(th) + 0.16666667f*__cosf(2.0f*th);
    float im = 0.66666667f*__sinf(th) + 0.16666667f*__sinf(2.0f*th);
    float d2 = re*re + im*im;
    return 1.0f/fmaxf(d2, 1e-7f);
}

// ---------------------------------------------------------------------------
__global__ __launch_bounds__(256) void init_kernel(float* grid, float* out) {
    int idx = blockIdx.x*256 + threadIdx.x;
    if (idx < KTOT) grid[idx] = 0.0f;
    if (idx == 0)   out[0] = 0.0f;
}

// ---------------------------------------------------------------------------
// Charge spreading: one thread per atom, 4x4x4 atomic scatter (hits L2)
__global__ __launch_bounds__(256) void spread_kernel(const float* __restrict__ pos,
                                                     const float* __restrict__ q,
                                                     const float* __restrict__ box,
                                                     float* __restrict__ grid) {
    int i = blockIdx.x*256 + threadIdx.x;
    if (i >= NATOMS) return;
    float bx[9], ib[9];
    #pragma unroll
    for (int r = 0; r < 9; ++r) bx[r] = box[r];
    inv3(bx, ib);
    float px = pos[i*3], py = pos[i*3+1], pz = pos[i*3+2];
    float fx = px*ib[0] + py*ib[3] + pz*ib[6];
    float fy = px*ib[1] + py*ib[4] + pz*ib[7];
    float fz = px*ib[2] + py*ib[5] + pz*ib[8];
    fx -= floorf(fx); fy -= floorf(fy); fz -= floorf(fz);
    float ux = fx*64.0f, uy = fy*64.0f, uz = fz*64.0f;
    int bxx = (int)floorf(ux), byy = (int)floorf(uy), bzz = (int)floorf(uz);
    float wx[4], wy[4], wz[4];
    bsp4(ux - floorf(ux), wx);
    bsp4(uy - floorf(uy), wy);
    bsp4(uz - floorf(uz), wz);
    float qi = q[i];
    #pragma unroll
    for (int jx = 0; jx < 4; ++jx) {
        int ix = (bxx + jx - 3) & 63;
        #pragma unroll
        for (int jy = 0; jy < 4; ++jy) {
            int iy = (byy + jy - 3) & 63;
            float wxy = qi*wx[jx]*wy[jy];
            #pragma unroll
            for (int jz = 0; jz < 4; ++jz) {
                int iz = (bzz + jz - 3) & 63;
                atomicAdd(&grid[(ix*64 + iy)*64 + iz], wxy*wz[jz]);
            }
        }
    }
}

// ---------------------------------------------------------------------------
// One 1D DFT pass as a 64x64 x 4096-line complex GEMM via WMMA f32 16x16x4.
// Twiddle matrix W[m][z] = e^{-2pi i m z / 64} built in LDS per block.
// A fragment (16x4 f32): lanes 0-15 hold K=0,1; lanes 16-31 hold K=2,3.
// C/D (16x16 f32): 8 VGPRs, lane half selects rows M=0-7 / M=8-15.
// Element X[k][n] addressed as in + map(n) + k*sK ; output at map(n) + m*sM.
__global__ __launch_bounds__(128) void dft_pass(const float* __restrict__ inR,
                                                const float* __restrict__ inI,
                                                float* __restrict__ outR,
                                                float* __restrict__ outI,
                                                int nDiv, int nMulHi, int nMulLo,
                                                int sK, int sM, int complexIn) {
    __shared__ float wr[64*64];
    __shared__ float wi[64*64];
    int tid = threadIdx.x;
    for (int idx = tid; idx < 64*64; idx += 128) {
        int m = idx >> 6, z = idx & 63;
        int p = (m * z) & 63;                       // phase mod 64 exactly
        float ang = -(PI_F/32.0f)*(float)p;         // e^{-i*2pi*mz/64}
        wr[idx] = __cosf(ang);
        wi[idx] = __sinf(ang);
    }
    __syncthreads();

    int wave = tid >> 5;
    int lane = tid & 31;
    int tile = blockIdx.x*4 + wave;                 // 1024 tiles total
    int mbase = (tile & 3)*16;                      // 4 m-tiles
    int nbase = (tile >> 2)*16;                     // 256 n-tiles
    int half = lane >> 4;
    int lm   = lane & 15;
    int n    = nbase + lm;
    int mapn = (n / nDiv)*nMulHi + (n % nDiv)*nMulLo;

    v8f cr = {}; v8f ci = {};
    for (int k0 = 0; k0 < 64; k0 += 4) {
        int k = k0 + 2*half;
        v2f ar, ai, br;
        ar.x = wr[(mbase + lm)*64 + k];
        ar.y = wr[(mbase + lm)*64 + k + 1];
        ai.x = wi[(mbase + lm)*64 + k];
        ai.y = wi[(mbase + lm)*64 + k + 1];
        br.x = inR[mapn + k*sK];
        br.y = inR[mapn + (k + 1)*sK];
        // Cr += Wr*Xr ; Ci += Wi*Xr
        cr = __builtin_amdgcn_wmma_f32_16x16x4_f32(false, ar, false, br, (short)0, cr, false, false);
        ci = __builtin_amdgcn_wmma_f32_16x16x4_f32(false, ai, false, br, (short)0, ci, false, false);
        if (complexIn) {
            v2f bi, ain;
            bi.x = inI[mapn + k*sK];
            bi.y = inI[mapn + (k + 1)*sK];
            ain.x = -ai.x; ain.y = -ai.y;           // f32 WMMA has no A-neg modifier
            // Cr += (-Wi)*Xi ; Ci += Wr*Xi
            cr = __builtin_amdgcn_wmma_f32_16x16x4_f32(false, ain, false, bi, (short)0, cr, false, false);
            ci = __builtin_amdgcn_wmma_f32_16x16x4_f32(false, ar, false, bi, (short)0, ci, false, false);
        }
    }
    #pragma unroll
    for (int v = 0; v < 8; ++v) {
        int m = mbase + v + 8*half;
        outR[mapn + m*sM] = cr[v];
        outI[mapn + m*sM] = ci[v];
    }
}

// ---------------------------------------------------------------------------
// Reciprocal-space energy reduction over all 64^3 modes
__global__ __launch_bounds__(256) void recip_kernel(const float* __restrict__ Fr,
                                                    const float* __restrict__ Fi,
                                                    const float* __restrict__ box,
                                                    float* __restrict__ out) {
    __shared__ float red[256];
    int idx = blockIdx.x*256 + threadIdx.x;
    int kx = idx >> 12, ky = (idx >> 6) & 63, kz = idx & 63;
    float bx[9], ib[9];
    #pragma unroll
    for (int r = 0; r < 9; ++r) bx[r] = box[r];
    inv3(bx, ib);
    float mx = (kx < 32) ? (float)kx : (float)(kx - 64);
    float my = (ky < 32) ? (float)ky : (float)(ky - 64);
    float mz = (kz < 32) ? (float)kz : (float)(kz - 64);
    // kvec[a] = sum_b m[b] * inv_box[a][b]
    float kv0 = mx*ib[0] + my*ib[1] + mz*ib[2];
    float kv1 = mx*ib[3] + my*ib[4] + mz*ib[5];
    float kv2 = mx*ib[6] + my*ib[7] + mz*ib[8];
    float msq = kv0*kv0 + kv1*kv1 + kv2*kv2;
    float term = 0.0f;
    if (msq > 0.0f) {
        float infl = __expf(-(PI_F*PI_F)*msq/(ALPHA_C*ALPHA_C)) / msq;
        float b3   = bmodf(kx)*bmodf(ky)*bmodf(kz);
        float fr = Fr[idx], fi = Fi[idx];
        term = infl*b3*(fr*fr + fi*fi);
    }
    red[threadIdx.x] = term;
    __syncthreads();
    for (int s = 128; s > 0; s >>= 1) {
        if (threadIdx.x < s) red[threadIdx.x] += red[threadIdx.x + s];
        __syncthreads();
    }
    if (threadIdx.x == 0) {
        float V = fabsf(det3(bx));
        atomicAdd(out, (COULOMB_C/(2.0f*PI_F*V))*red[0]);
    }
}

// ---------------------------------------------------------------------------
// Direct space (tiled all-pairs through LDS) + self energy + exclusion corr.
// grid = (16 i-chunks, 4 j-chunks); self/exclusion folded into j-chunk 0.
__global__ __launch_bounds__(256) void direct_kernel(const float* __restrict__ pos,
                                                     const float* __restrict__ q,
                                                     const float* __restrict__ box,
                                                     const int*  __restrict__ excl,
                                                     float* __restrict__ out) {
    __shared__ float sh[256*4];
    __shared__ float red[256];
    float bx[9], ib[9];
    #pragma unroll
    for (int r = 0; r < 9; ++r) bx[r] = box[r];
    inv3(bx, ib);

    int i  = blockIdx.x*256 + threadIdx.x;
    float pix = pos[i*3], piy = pos[i*3+1], piz = pos[i*3+2];
    float qi  = q[i];
    int e0 = excl[i*2], e1 = excl[i*2+1];

    int j0 = blockIdx.y*1024;                       // this block's j-range
    float acc = 0.0f;
    for (int tile0 = j0; tile0 < j0 + 1024; tile0 += 256) {
        int j = tile0 + threadIdx.x;
        sh[threadIdx.x*4+0] = pos[j*3];
        sh[threadIdx.x*4+1] = pos[j*3+1];
        sh[threadIdx.x*4+2] = pos[j*3+2];
        sh[threadIdx.x*4+3] = q[j];
        __syncthreads();
        for (int jj = 0; jj < 256; ++jj) {
            int jg = tile0 + jj;
            float dx = pix - sh[jj*4+0];
            float dy = piy - sh[jj*4+1];
            float dz = piz - sh[jj*4+2];
            min_image(dx, dy, dz, bx, ib);
            float r2 = dx*dx + dy*dy + dz*dz;
            if (r2 < CUT2 && jg != i && jg != e0 && jg != e1) {
                float r = sqrtf(r2);
                acc += qi*sh[jj*4+3]*erfcf(ALPHA_C*r)/r;
            }
        }
        __syncthreads();
    }
    float e = 0.5f*COULOMB_C*acc;                   // ordered pairs -> x0.5

    if (blockIdx.y == 0) {
        // self energy
        e += -COULOMB_C*ALPHA_C*0.5641895835477563f*qi*qi;  // 1/sqrt(pi)
        // exclusion correction (reciprocal-space removal)
        #pragma unroll
        for (int t = 0; t < 2; ++t) {
            int ej = (t == 0) ? e0 : e1;
            if (ej >= 0) {
                float dx = pix - pos[ej*3];
                float dy = piy - pos[ej*3+1];
                float dz = piz - pos[ej*3+2];
                min_image(dx, dy, dz, bx, ib);
                float re2 = dx*dx + dy*dy + dz*dz;
                if (re2 > 0.0f) {
                    float r = sqrtf(re2);
                    e += -0.5f*COULOMB_C*qi*q[ej]*erff(ALPHA_C*r)/r;
                }
            }
        }
    }

    red[threadIdx.x] = e;
    __syncthreads();
    for (int s = 128; s > 0; s >>= 1) {
        if (threadIdx.x < s) red[threadIdx.x] += red[threadIdx.x + s];
        __syncthreads();
    }
    if (threadIdx.x == 0) atomicAdd(out, red[0]);
}

// ---------------------------------------------------------------------------
extern "C" void kernel_launch(void* const* d_in, const int* in_sizes, int n_in,
                              void* d_out, int out_size, void* d_ws, size_t ws_size,
                              hipStream_t stream) {
    const float* pos  = (const float*)d_in[0];
    const float* q    = (const float*)d_in[1];
    const float* box  = (const float*)d_in[2];
    const int*   excl = (const int*)d_in[3];
    float* out = (float*)d_out;
    float* ws  = (float*)d_ws;

    float* grid = ws;                // 64^3 real charge grid       (1 MB)
    float* F1r  = ws + 1*KTOT;       // complex plane after pass Z / final
    float* F1i  = ws + 2*KTOT;
    float* F2r  = ws + 3*KTOT;       // complex plane after pass Y
    float* F2i  = ws + 4*KTOT;       // total workspace: 5 MB

    init_kernel  <<<KTOT/256, 256, 0, stream>>>(grid, out);
    spread_kernel<<<NATOMS/256, 256, 0, stream>>>(pos, q, box, grid);

    // 3 axis DFT passes: 1024 tiles of 16x16, 4 waves/block -> 256 blocks
    // layout (x*64+y)*64+z ; map(n) = (n/nDiv)*nMulHi + (n%nDiv)*nMulLo
    // pass Z: lines=(x,y), K along z (stride 1),  out kz stride 1
    dft_pass<<<256, 128, 0, stream>>>(grid, nullptr, F1r, F1i,
                                      4096, 0, 64,   1,    1,    0);
    // pass Y: lines=(x,kz), K along y (stride 64), out ky stride 64
    dft_pass<<<256, 128, 0, stream>>>(F1r, F1i, F2r, F2i,
                                      64, 4096, 1,   64,   64,   1);
    // pass X: lines=(ky,kz), K along x (stride 4096), out kx stride 4096
    dft_pass<<<256, 128, 0, stream>>>(F2r, F2i, F1r, F1i,
                                      4096, 0, 1,    4096, 4096, 1);

    recip_kernel <<<KTOT/256, 256, 0, stream>>>(F1r, F1i, box, out);

    dim3 dgrid(NATOMS/256, 4);
    direct_kernel<<<dgrid, 256, 0, stream>>>(pos, q, box, excl, out);
}